// DWTExtractor_24215025615116
// MI455X (gfx1250) — compile-verified
//
#include <hip/hip_runtime.h>

// ---------------------------------------------------------------------------
// Fused 2-level Haar DWT + bilinear-upsample feature extractor for MI455X.
// Bandwidth-bound: ~320 MiB @ 23.3 TB/s => ~14 us floor; arithmetic is ~1-2
// GFLOP total, so matrix ops are irrelevant. CDNA5 features used:
//   - global_load_async_to_lds_b128 (ASYNCcnt) to stage input tiles into LDS
//   - s_wait_asynccnt + workgroup barrier pipeline
//   - non-temporal streaming stores for the write-once 192 MiB output
// Tile sized for occupancy: 30.2 KB LDS/block -> ~10 blocks per 320 KB WGP,
// so many workgroups overlap their async-load/wait/barrier phases.
// ---------------------------------------------------------------------------

#define TILE      32          // output tile edge (level-1 / output space)
#define IN_T      72          // input tile edge incl. 4-px halo (2*TILE + 8)
#define IN_STRIDE 72          // 288 B rows: keeps b128 LDS writes 16-B aligned
#define A_T       36          // cA1 tile incl. halo (TILE + 4)
#define A_STRIDE  37          // padded
#define H_T       18          // level-2 tile incl. 1-elt halo (TILE/2 + 2)
#define H_STRIDE  19          // padded

// LDS layout (float units)
#define OFF_IN 0
#define OFF_A  (IN_T * IN_STRIDE)                  // 5184
#define OFF_H  (OFF_A + A_T * A_STRIDE)            // +1332
#define OFF_V  (OFF_H + H_T * H_STRIDE)            // +342
#define OFF_D  (OFF_V + H_T * H_STRIDE)            // +342
#define SMEM_FLOATS (OFF_D + H_T * H_STRIDE)       // 7542 floats = 30168 B

#define PLANE  262144  // 512*512

__global__ __launch_bounds__(256)
void dwt2_haar_fused_kernel(const float* __restrict__ in, float* __restrict__ out)
{
    extern __shared__ float smem[];
    float* sIn = smem + OFF_IN;
    float* sA  = smem + OFF_A;
    float* sH  = smem + OFF_H;
    float* sV  = smem + OFF_V;
    float* sD  = smem + OFF_D;

    const int tid = threadIdx.x;
    const int tX  = blockIdx.x;
    const int tY  = blockIdx.y;
    const int b   = blockIdx.z;

    const int oy0 = TILE * tY;            // output / level-1 coords
    const int ox0 = TILE * tX;
    const int k0  = (TILE / 2) * tY;      // level-2 coords
    const int l0  = (TILE / 2) * tX;
    const int iy0 = 4 * k0 - 4;           // input row of sIn[0][0]
    const int ix0 = 4 * l0 - 4;

    const float* inB = in + ((size_t)b << 20);   // b * 1024 * 1024

    // ---------------- Phase 0: async-stage input tile into LDS -------------
    // 72 rows x 18 float4-chunks; addresses clamped in-bounds. Clamped halo
    // data is garbage but provably unused (bilinear edge clamp renormalizes
    // to pure boundary replication, so out-of-image taps are never read).
    {
        const unsigned sIn_base =
            (unsigned)(unsigned long long)(size_t)sIn;  // flat->LDS offset = addr[31:0]
        for (int q = tid; q < IN_T * (IN_T / 4); q += 256) {
            int row = q / (IN_T / 4);
            int cc  = q - row * (IN_T / 4);
            int gr = iy0 + row;     gr = gr < 0 ? 0 : (gr > 1023 ? 1023 : gr);
            int gc = ix0 + 4 * cc;  gc = gc < 0 ? 0 : (gc > 1020 ? 1020 : gc);
            const float* gp = inB + (size_t)gr * 1024 + gc;
            unsigned loff = sIn_base + (unsigned)((row * IN_STRIDE + 4 * cc) * 4);
            asm volatile("global_load_async_to_lds_b128 %0, %1, off"
                         :: "v"(loff), "v"(gp) : "memory");
        }
        asm volatile("s_wait_asynccnt 0" ::: "memory");
    }
    __syncthreads();

    // ---------------- Phase 1: level-1 Haar; cA1 -> LDS, details -> out ----
    for (int t = tid; t < A_T * A_T; t += 256) {
        int r = t / A_T, c = t - r * A_T;
        const float* p = &sIn[(2 * r) * IN_STRIDE + 2 * c];
        float x00 = p[0], x01 = p[1];
        float x10 = p[IN_STRIDE], x11 = p[IN_STRIDE + 1];
        sA[r * A_STRIDE + c] = (x00 + x01 + x10 + x11) * 0.5f;
        if (r >= 2 && r < 2 + TILE && c >= 2 && c < 2 + TILE) {
            int i = oy0 + r - 2, j = ox0 + c - 2;
            size_t base = (((size_t)b * 6 + 0) * 512 + i) * 512 + j;
            __builtin_nontemporal_store((x00 - x01 + x10 - x11) * 0.5f, out + base);             // cH1
            __builtin_nontemporal_store((x00 + x01 - x10 - x11) * 0.5f, out + base + PLANE);     // cV1
            __builtin_nontemporal_store((x00 - x01 - x10 + x11) * 0.5f, out + base + 2 * PLANE); // cD1
        }
    }
    __syncthreads();

    // ---------------- Phase 2: level-2 Haar on cA1 tile --------------------
    for (int t = tid; t < H_T * H_T; t += 256) {
        int r = t / H_T, c = t - r * H_T;
        const float* p = &sA[(2 * r) * A_STRIDE + 2 * c];
        float a00 = p[0], a01 = p[1];
        float a10 = p[A_STRIDE], a11 = p[A_STRIDE + 1];
        int o = r * H_STRIDE + c;
        sH[o] = (a00 - a01 + a10 - a11) * 0.5f;
        sV[o] = (a00 + a01 - a10 - a11) * 0.5f;
        sD[o] = (a00 - a01 - a10 + a11) * 0.5f;
    }
    __syncthreads();

    // ---------------- Phase 3: half-pixel bilinear 2x upsample -------------
    // src = dst/2 - 0.25: even dst -> 0.25*in[m-1] + 0.75*in[m],
    //                     odd  dst -> 0.75*in[m]   + 0.25*in[m+1], edge-clamped.
    for (int t = tid; t < TILE * TILE; t += 256) {
        int dy = t >> 5, dx = t & (TILE - 1);

        int hy = dy >> 1, r0, r1; float wy0, wy1;
        if (dy & 1) { r0 = hy + 1; r1 = hy + 2; wy0 = 0.75f; wy1 = 0.25f; }
        else        { r0 = hy;     r1 = hy + 1; wy0 = 0.25f; wy1 = 0.75f; }
        int kA = k0 - 1 + r0; kA = kA < 0 ? 0 : (kA > 255 ? 255 : kA); r0 = kA - k0 + 1;
        int kB = k0 - 1 + r1; kB = kB < 0 ? 0 : (kB > 255 ? 255 : kB); r1 = kB - k0 + 1;

        int hx = dx >> 1, c0, c1; float wx0, wx1;
        if (dx & 1) { c0 = hx + 1; c1 = hx + 2; wx0 = 0.75f; wx1 = 0.25f; }
        else        { c0 = hx;     c1 = hx + 1; wx0 = 0.25f; wx1 = 0.75f; }
        int lA = l0 - 1 + c0; lA = lA < 0 ? 0 : (lA > 255 ? 255 : lA); c0 = lA - l0 + 1;
        int lB = l0 - 1 + c1; lB = lB < 0 ? 0 : (lB > 255 ? 255 : lB); c1 = lB - l0 + 1;

        int i00 = r0 * H_STRIDE + c0, i01 = r0 * H_STRIDE + c1;
        int i10 = r1 * H_STRIDE + c0, i11 = r1 * H_STRIDE + c1;
        float w00 = wy0 * wx0, w01 = wy0 * wx1, w10 = wy1 * wx0, w11 = wy1 * wx1;

        int oy = oy0 + dy, ox = ox0 + dx;
        size_t base = (((size_t)b * 6 + 3) * 512 + oy) * 512 + ox;
        __builtin_nontemporal_store(
            w00 * sH[i00] + w01 * sH[i01] + w10 * sH[i10] + w11 * sH[i11], out + base);
        __builtin_nontemporal_store(
            w00 * sV[i00] + w01 * sV[i01] + w10 * sV[i10] + w11 * sV[i11], out + base + PLANE);
        __builtin_nontemporal_store(
            w00 * sD[i00] + w01 * sD[i01] + w10 * sD[i10] + w11 * sD[i11], out + base + 2 * PLANE);
    }
}

extern "C" void kernel_launch(void* const* d_in, const int* in_sizes, int n_in,
                              void* d_out, int out_size, void* d_ws, size_t ws_size,
                              hipStream_t stream)
{
    const float* x = (const float*)d_in[0];
    float* out = (float*)d_out;
    int B = in_sizes[0] >> 20;                 // elems / (1024*1024)
    dim3 grid(512 / TILE, 512 / TILE, B);      // (16, 16, B)
    dim3 block(256);                           // 8 wave32s
    size_t smem = (size_t)SMEM_FLOATS * sizeof(float);  // ~30.2 KB
    hipLaunchKernelGGL(dwt2_haar_fused_kernel, grid, block, smem, stream, x, out);
}